// KNNSelfLayer_53712861004190
// MI455X (gfx1250) — compile-verified
//
#include <hip/hip_runtime.h>
#include <stdint.h>

#define BATCH   2
#define NPTS    1024
#define FDIM    64
#define KSEL    17          // K_NEIGHBORS + 1
#define PTS     8           // points per workgroup == waves per block
#define JCH     32          // candidate rows per chunk (double-buffered)
#define ROWPAD  68          // floats per padded LDS row (17 x float4) -> bank-conflict relief
#define THREADS 256         // 8 wave32 waves

typedef int cdna5_v4i __attribute__((ext_vector_type(4)));

// ---- CDNA5 async global->LDS copy (gfx1250) -------------------------------
#if defined(__has_builtin)
# if __has_builtin(__builtin_amdgcn_global_load_async_to_lds_b128)
#  define ASYNC_B128(g, l)                                                     \
     __builtin_amdgcn_global_load_async_to_lds_b128(                           \
         (__attribute__((address_space(1))) cdna5_v4i*)(void*)(g),             \
         (__attribute__((address_space(3))) cdna5_v4i*)(void*)(l), 0, 0)
#  define HAVE_ASYNC_BUILTIN 1
# endif
# if __has_builtin(__builtin_amdgcn_s_wait_asynccnt)
#  define WAIT_ASYNC() __builtin_amdgcn_s_wait_asynccnt(0)
#  define HAVE_WAIT_BUILTIN 1
# endif
#endif
#ifndef HAVE_ASYNC_BUILTIN
#  define ASYNC_B128(g, l)                                                     \
     asm volatile("global_load_async_to_lds_b128 %0, %1, off"                  \
                  :: "v"((uint32_t)(uintptr_t)                                 \
                         (__attribute__((address_space(3))) void*)(l)),        \
                     "v"((const void*)(g))                                     \
                  : "memory")
#endif
#ifndef HAVE_WAIT_BUILTIN
#  define WAIT_ASYNC() asm volatile("s_wait_asynccnt 0x0" ::: "memory")
#endif
// ---------------------------------------------------------------------------

__global__ __launch_bounds__(THREADS)
void knn_l1_topk_gather(const float* __restrict__ x, float* __restrict__ out) {
    __shared__ float xj[2][JCH * ROWPAD];  // ping-pong candidate chunks (2 x 8.5 KB)
    __shared__ float dist[PTS * NPTS];     // full distance rows for this block (32 KB)
    __shared__ int   sel[PTS * KSEL];      // selected neighbor indices

    const int t    = threadIdx.x;
    const int wave = t >> 5;               // 8 waves == PTS points
    const int lane = t & 31;
    const int wg   = blockIdx.x;           // 0 .. B*(N/PTS)-1
    const int b    = wg / (NPTS / PTS);
    const int p0   = (wg % (NPTS / PTS)) * PTS;
    const float* __restrict__ xb = x + (size_t)b * NPTS * FDIM;

    // ---- query row is wave-uniform: keep it in 16 float4 registers ----
    float4 arow[FDIM / 4];
    {
        const float4* __restrict__ ar =
            (const float4*)(xb + (size_t)(p0 + wave) * FDIM);
#pragma unroll
        for (int c = 0; c < FDIM / 4; ++c) arow[c] = ar[c];
    }

    // helper: issue one chunk's async loads (JCH*16 = 512 b128 units, 2/thread)
    auto issue_chunk = [&](int ch, int buf) {
#pragma unroll
        for (int u = 0; u < (JCH * (FDIM / 4)) / THREADS; ++u) {
            const int unit = t + u * THREADS;
            const int r = unit >> 4, c = unit & 15;
            ASYNC_B128(xb + (size_t)(ch * JCH + r) * FDIM + c * 4,
                       &xj[buf][r * ROWPAD + c * 4]);
        }
    };

    // ---- software-pipelined distance phase: 32 chunks of 32 rows ----
    issue_chunk(0, 0);
    WAIT_ASYNC();
    __syncthreads();

    for (int ch = 0; ch < NPTS / JCH; ++ch) {
        const int cur = ch & 1;
        if (ch + 1 < NPTS / JCH) issue_chunk(ch + 1, cur ^ 1);

        // wave `wave` vs candidate row `lane` of this chunk
        const float4* __restrict__ bj = (const float4*)&xj[cur][lane * ROWPAD];
        float acc = 0.0f;
#pragma unroll
        for (int c = 0; c < FDIM / 4; ++c) {
            const float4 a = arow[c];
            const float4 q = bj[c];
            acc += fabsf(a.x - q.x) + fabsf(a.y - q.y)
                 + fabsf(a.z - q.z) + fabsf(a.w - q.w);
        }
        dist[wave * NPTS + ch * JCH + lane] = acc;

        WAIT_ASYNC();      // next chunk's async loads (this wave's share) done
        __syncthreads();   // all reads of cur finished + all waves' loads landed
    }

    // ---- top-17 selection: wave w owns point (p0 + w) ----
    float* __restrict__ drow = &dist[wave * NPTS];
    for (int k = 0; k < KSEL; ++k) {
        // key = (dist_bits << 32) | j : dist >= 0 so float bits are monotone;
        // min-key == smallest distance, ties broken by lowest index (matches top_k)
        unsigned long long best = ~0ull;
        for (int j = lane; j < NPTS; j += 32) {
            const unsigned long long key =
                ((unsigned long long)__float_as_uint(drow[j]) << 32) | (unsigned)j;
            best = key < best ? key : best;
        }
#pragma unroll
        for (int off = 16; off > 0; off >>= 1) {
            const unsigned long long o = __shfl_xor(best, off, 32);
            best = o < best ? o : best;
        }
        const int jmin = (int)(best & 0xffffffffu);
        if (lane == 0) {
            sel[wave * KSEL + k] = jmin;
            drow[jmin] = __uint_as_float(0x7f800000u);   // +inf: exclude next passes
        }
        __syncthreads();
    }

    // ---- gather + transpose: out[b][p][f][k] = x[b][sel[k]][f] ----
    float* __restrict__ op =
        out + (size_t)(b * NPTS + (p0 + wave)) * FDIM * KSEL;
    for (int e = lane; e < FDIM * KSEL; e += 32) {
        const int f  = e / KSEL;
        const int kk = e % KSEL;
        const int j  = sel[wave * KSEL + kk];
        op[e] = xb[(size_t)j * FDIM + f];
    }
}

extern "C" void kernel_launch(void* const* d_in, const int* in_sizes, int n_in,
                              void* d_out, int out_size, void* d_ws, size_t ws_size,
                              hipStream_t stream) {
    (void)in_sizes; (void)n_in; (void)d_ws; (void)ws_size; (void)out_size;
    const float* x = (const float*)d_in[0];
    float* out = (float*)d_out;
    dim3 grid(BATCH * (NPTS / PTS));   // 256 workgroups
    knn_l1_topk_gather<<<grid, THREADS, 0, stream>>>(x, out);
}